// Sampler_22050362098045
// MI455X (gfx1250) — compile-verified
//
#include <hip/hip_runtime.h>
#include <stdint.h>

#define BATCH 2048
#define VOCAB 50257

// ---------------------------------------------------------------------------
// Kernel 1: DMA memset of d_out via the Tensor Data Mover (CDNA5 TDM).
// Each single-wave workgroup zeroes a 16 KB LDS staging buffer once, then
// issues TENSOR_STORE_FROM_LDS descriptors streaming those zeros to global
// memory. Tiles are exactly sized (last tile smaller) so no byte is written
// past d_out. Completion tracked with TENSORcnt.
// ---------------------------------------------------------------------------

typedef unsigned int v4u  __attribute__((ext_vector_type(4)));
typedef int          v8i_t __attribute__((ext_vector_type(8)));
typedef int          v4i_t __attribute__((ext_vector_type(4)));

#define TILE_UNITS 2048u                 // 8-byte units per TDM tile
#define TILE_BYTES (TILE_UNITS * 8u)     // 16 KB

__global__ __launch_bounds__(32) void tdm_zero_fill(float* __restrict__ out,
                                                    unsigned int nTiles,
                                                    unsigned int lastTileUnits) {
    __shared__ unsigned long long zbuf[TILE_UNITS];  // 16 KB of zeros (LDS offset 0)

    // Zero the LDS staging buffer (one wave, 512 B per lane).
    for (unsigned int i = threadIdx.x; i < TILE_UNITS; i += 32u) {
        zbuf[i] = 0ull;
    }
    // Split-counter wait: LDS writes must land before the TDM reads LDS
    // (TDM ops are unordered vs. DS ops).
    asm volatile("s_wait_dscnt 0x0" ::: "memory");

    const unsigned long long base = (unsigned long long)(uintptr_t)out;

    for (unsigned int t = blockIdx.x; t < nTiles; t += gridDim.x) {
        const unsigned long long ga = base + (unsigned long long)t * TILE_BYTES;
        const unsigned int tileUnits = (t == nTiles - 1u) ? lastTileUnits : TILE_UNITS;

        // ---- D# group 0 (ISA 8.3): count=1 | lds_addr=0 | global_addr | type=2
        v4u g0;
        g0.x = 0x1u;                                        // count=1, user mode
        g0.y = 0u;                                          // lds_addr = 0 (zbuf)
        g0.z = (unsigned int)(ga & 0xFFFFFFFFull);          // global_addr[31:0]
        g0.w = (unsigned int)((ga >> 32) & 0x01FFFFFFull)   // global_addr[56:32]
             | (2u << 30);                                  // type = 2 ("image")

        // ---- D# group 1 (ISA 8.4): data_size=8B, 1-D tile of tileUnits units
        const unsigned int td0 = 0x7FFFFFFFu;               // huge tensor_dim0 (exact clip via tile size)
        v8i_t g1;
        g1[0] = (int)(3u << 16);                            // data_size=3 (8B); no iterate/pad/multicast
        g1[1] = (int)((td0 & 0xFFFFu) << 16);               // tensor_dim0[15:0] @ bits 63:48
        g1[2] = (int)(((td0 >> 16) & 0xFFFFu)               // tensor_dim0[31:16] @ bits 79:64
             | (1u << 16));                                 // tensor_dim1 = 1 @ bits 95:80
        g1[3] = (int)(tileUnits << 16);                     // tile_dim0 @ bits 127:112
        g1[4] = 0;                                          // tile_dim1=0 (1-D), tile_dim2=0
        g1[5] = (int)td0;                                   // tensor_dim0_stride[31:0]
        g1[6] = 0;                                          // stride0 hi / stride1 lo
        g1[7] = 0;                                          // stride1 hi

        // ---- D# group 2 (ISA 8.5): no iteration; higher dims = 1/unused
        v4i_t g2;
        g2[0] = 1;                                          // tensor_dim2
        g2[1] = 1;                                          // tensor_dim3
        g2[2] = 0;                                          // tensor_dim2_stride lo
        g2[3] = 0;                                          // stride hi | tile_dim3=0

        // ---- D# group 3 (ISA 8.6): unused
        v4i_t g3 = {0, 0, 0, 0};

#if defined(__clang_major__) && (__clang_major__ >= 23)
        // amdgpu-toolchain (clang-23 / therock-10.0): 6-arg form
        v8i_t g4 = {0, 0, 0, 0, 0, 0, 0, 0};
        __builtin_amdgcn_tensor_store_from_lds(g0, g1, g2, g3, g4, 0);
#else
        // ROCm 7.2 (clang-22): 5-arg form
        __builtin_amdgcn_tensor_store_from_lds(g0, g1, g2, g3, 0);
#endif
    }

    // Drain all outstanding TDM transfers before wave end.
    __builtin_amdgcn_s_wait_tensorcnt(0);
}

// ---------------------------------------------------------------------------
// Kernel 2: one wave32 per row. Scan 128 elements/chunk (float4 per lane):
// lane-local prefix + 5-step shuffle wave scan, ballot for first crossing,
// write the single 1.0f, uniform early exit. Inputs are uniform[0,1) so the
// crossing is almost always in the first chunk -> ~1 MB total read traffic.
// ---------------------------------------------------------------------------
__global__ __launch_bounds__(256) void crossing_kernel(const float* __restrict__ in,
                                                       const float* __restrict__ u,
                                                       float* __restrict__ out) {
    const int lane = threadIdx.x & 31;
    const int row  = (blockIdx.x << 3) + (threadIdx.x >> 5);   // 8 waves/block

    const float  uv = u[row];
    const float* rp = in + (size_t)row * VOCAB;

    float running = 0.0f;
    for (int base = 0; base < VOCAB; base += 128) {
        // CDNA5 prefetch of the next chunk (global_prefetch_b8) for the rare
        // rows that need more than one chunk.
        if (base + 128 < VOCAB) {
            __builtin_prefetch(rp + base + 128 + (lane << 2), 0, 0);
        }

        const int i0 = base + (lane << 2);
        float4 v;
        if (i0 + 3 < VOCAB) {
            v = *reinterpret_cast<const float4*>(rp + i0);
        } else {
            v.x = (i0 + 0 < VOCAB) ? rp[i0 + 0] : 0.0f;
            v.y = (i0 + 1 < VOCAB) ? rp[i0 + 1] : 0.0f;
            v.z = (i0 + 2 < VOCAB) ? rp[i0 + 2] : 0.0f;
            v.w = (i0 + 3 < VOCAB) ? rp[i0 + 3] : 0.0f;
        }

        // Lane-local inclusive prefix over 4 elements.
        const float s0 = v.x;
        const float s1 = s0 + v.y;
        const float s2 = s1 + v.z;
        const float s3 = s2 + v.w;

        // Wave32 inclusive scan of per-lane sums (5 shuffle steps).
        float incl = s3;
        #pragma unroll
        for (int off = 1; off < 32; off <<= 1) {
            const float n = __shfl_up(incl, off, 32);
            if (lane >= off) incl += n;
        }
        const float excl = incl - s3;

        const float c0 = running + excl + s0;
        const float c1 = running + excl + s1;
        const float c2 = running + excl + s2;
        const float c3 = running + excl + s3;

        // Which lanes have reached/passed the threshold by their last element?
        const unsigned long long mask = __ballot(c3 >= uv);
        if (mask != 0ull) {
            const int firstlane = __ffsll(mask) - 1;
            if (lane == firstlane) {
                // Minimal k with c_k >= uv; increments are >= 0 so this is a
                // real (non-padded) element.
                const int k = (c0 >= uv) ? 0 : (c1 >= uv) ? 1 : (c2 >= uv) ? 2 : 3;
                out[(size_t)row * VOCAB + i0 + k] = 1.0f;
            }
            return;  // mask is wave-uniform: whole wave exits together
        }
        running += __shfl(incl, 31, 32);  // broadcast chunk total
    }
    // Never crossed: row stays all zeros (matches reference).
}

// ---------------------------------------------------------------------------
extern "C" void kernel_launch(void* const* d_in, const int* in_sizes, int n_in,
                              void* d_out, int out_size, void* d_ws, size_t ws_size,
                              hipStream_t stream) {
    (void)in_sizes; (void)n_in; (void)d_ws; (void)ws_size; (void)out_size;

    const float* inputs = (const float*)d_in[0];
    const float* uvals  = (const float*)d_in[1];
    float*       out    = (float*)d_out;

    // Exact tiling of the 412 MB output in 8-byte units.
    const unsigned long long totalBytes = (unsigned long long)BATCH * VOCAB * sizeof(float);
    const unsigned long long totalUnits = totalBytes / 8ull;                  // divisible: BATCH even
    const unsigned int nTiles =
        (unsigned int)((totalUnits + TILE_UNITS - 1ull) / TILE_UNITS);
    const unsigned int lastTileUnits =
        (unsigned int)(totalUnits - (unsigned long long)(nTiles - 1u) * TILE_UNITS);

    // 1) TDM-driven zero fill of the whole output.
    tdm_zero_fill<<<1024, 32, 0, stream>>>(out, nTiles, lastTileUnits);

    // 2) Per-row first-crossing one-hot writes (stream-ordered after the fill).
    crossing_kernel<<<BATCH / 8, 256, 0, stream>>>(inputs, uvals, out);
}